// KPlaneDensityField_10187662426218
// MI455X (gfx1250) — compile-verified
//
#include <hip/hip_runtime.h>

#define RES 256
#define FDIM 8
#define HID 64

typedef __attribute__((ext_vector_type(2))) float v2f;
typedef __attribute__((ext_vector_type(8))) float v8f;

// ---------------- small helpers ----------------
__device__ __forceinline__ int iclampi(int v, int lo, int hi) {
  return v < lo ? lo : (v > hi ? hi : v);
}
__device__ __forceinline__ float4 mul4s(float4 a, float s) {
  float4 r; r.x = a.x * s; r.y = a.y * s; r.z = a.z * s; r.w = a.w * s; return r;
}
__device__ __forceinline__ float4 fma4s(float4 a, float s, float4 c) {
  float4 r;
  r.x = fmaf(a.x, s, c.x); r.y = fmaf(a.y, s, c.y);
  r.z = fmaf(a.z, s, c.z); r.w = fmaf(a.w, s, c.w);
  return r;
}
__device__ __forceinline__ float4 mul44(float4 a, float4 b) {
  float4 r; r.x = a.x * b.x; r.y = a.y * b.y; r.z = a.z * b.z; r.w = a.w * b.w; return r;
}
// pick element s (0..7) from a v8f that is uniform across a 16-lane half
__device__ __forceinline__ float pick8(v8f c, int s) {
  float a01 = (s & 1) ? c[1] : c[0];
  float a23 = (s & 1) ? c[3] : c[2];
  float a45 = (s & 1) ? c[5] : c[4];
  float a67 = (s & 1) ? c[7] : c[6];
  float b0 = (s & 2) ? a23 : a01;
  float b1 = (s & 2) ? a67 : a45;
  return (s & 4) ? b1 : b0;
}

// ---------------- kernel 0: v = w2 @ w1  (8 floats) ----------------
__global__ void kplane_compute_v(const float* __restrict__ w1,  // [HID, FDIM]
                                 const float* __restrict__ w2,  // [1, HID]
                                 float* __restrict__ v) {
  int f = threadIdx.x;
  if (f < FDIM) {
    float acc = 0.0f;
    for (int h = 0; h < HID; ++h) acc = fmaf(w2[h], w1[h * FDIM + f], acc);
    v[f] = acc;
  }
}

// ---------------- kernel 1: repack [C,H,W] -> [H,W,C]; fold v into plane 0 ----
__global__ void __launch_bounds__(256)
kplane_repack(const float* __restrict__ pxy, const float* __restrict__ pxz,
              const float* __restrict__ pyz, const float* __restrict__ v,
              float* __restrict__ dst) {  // dst: 3 x [RES*RES][FDIM]
  int tid = blockIdx.x * 256 + threadIdx.x;   // 3 * 65536 jobs
  int plane = tid >> 16;
  int pix = tid & 0xFFFF;
  const float* src = (plane == 0) ? pxy : ((plane == 1) ? pxz : pyz);
  float* out = dst + (size_t)plane * (RES * RES * FDIM) + (size_t)pix * FDIM;
#pragma unroll
  for (int f = 0; f < FDIM; ++f) {
    float s = (plane == 0) ? v[f] : 1.0f;
    out[f] = src[(size_t)f * (RES * RES) + pix] * s;
  }
}

// ---------------- bilerp of one repacked plane (8 channels) ----------------
__device__ __forceinline__ void bilerp8(const float* __restrict__ P, float px,
                                        float py, float4& lo, float4& hi) {
  // px, py in [-1, 1]; align_corners=True
  float fx = (px + 1.0f) * (0.5f * (RES - 1));
  float fy = (py + 1.0f) * (0.5f * (RES - 1));
  float x0f = floorf(fx), y0f = floorf(fy);
  float wx = fx - x0f, wy = fy - y0f;
  int x0 = iclampi((int)x0f, 0, RES - 1);
  int y0 = iclampi((int)y0f, 0, RES - 1);
  int x1 = (x0 + 1 > RES - 1) ? RES - 1 : x0 + 1;
  int y1 = (y0 + 1 > RES - 1) ? RES - 1 : y0 + 1;
  const float4* r00 = (const float4*)(P + (size_t)(y0 * RES + x0) * FDIM);
  const float4* r01 = (const float4*)(P + (size_t)(y0 * RES + x1) * FDIM);
  const float4* r10 = (const float4*)(P + (size_t)(y1 * RES + x0) * FDIM);
  const float4* r11 = (const float4*)(P + (size_t)(y1 * RES + x1) * FDIM);
  float w00 = (1.0f - wx) * (1.0f - wy);
  float w01 = wx * (1.0f - wy);
  float w10 = (1.0f - wx) * wy;
  float w11 = wx * wy;
  float4 l = mul4s(r00[0], w00);
  l = fma4s(r01[0], w01, l);
  l = fma4s(r10[0], w10, l);
  l = fma4s(r11[0], w11, l);
  float4 h = mul4s(r00[1], w00);
  h = fma4s(r01[1], w01, h);
  h = fma4s(r10[1], w10, h);
  h = fma4s(r11[1], w11, h);
  lo = l;
  hi = h;
}

// ---------------- kernel 2: main gather + WMMA channel-reduction ------------
__global__ void __launch_bounds__(256)
kplane_main(const float* __restrict__ pts,     // [N,3]
            const float* __restrict__ planes,  // repacked, v folded into plane 0
            const float* __restrict__ aabb,    // [2,3]
            float* __restrict__ out) {
  int id = blockIdx.x * 256 + threadIdx.x;

  float p0 = pts[3 * id + 0];
  float p1 = pts[3 * id + 1];
  float p2 = pts[3 * id + 2];
  float lo0 = aabb[0], lo1 = aabb[1], lo2 = aabb[2];
  float hi0 = aabb[3], hi1 = aabb[4], hi2 = aabb[5];
  p0 = (p0 - lo0) * (2.0f / (hi0 - lo0)) - 1.0f;
  p1 = (p1 - lo1) * (2.0f / (hi1 - lo1)) - 1.0f;
  p2 = (p2 - lo2) * (2.0f / (hi2 - lo2)) - 1.0f;

  const float* PA = planes;                         // xy (pre-scaled by v)
  const float* PB = planes + RES * RES * FDIM;      // xz
  const float* PC = planes + 2 * RES * RES * FDIM;  // yz

  float4 glo, ghi, tlo, thi;
  bilerp8(PA, p0, p1, glo, ghi);
  bilerp8(PB, p0, p2, tlo, thi);
  glo = mul44(glo, tlo);
  ghi = mul44(ghi, thi);
  bilerp8(PC, p1, p2, tlo, thi);
  glo = mul44(glo, tlo);
  ghi = mul44(ghi, thi);

  float g0 = glo.x, g1 = glo.y, g2 = glo.z, g3 = glo.w;
  float g4 = ghi.x, g5 = ghi.y, g6 = ghi.z, g7 = ghi.w;

  // sigma_m = sum_f g_f(point m).  Use V_WMMA_F32_16X16X4_F32 with B == ones:
  // D[m,n] = sum_k A[m,k].  A layout (32-bit 16x4): lane m holds K=0,1;
  // lane m+16 holds K=2,3 for point m -> assemble with shuffles + selects
  // (no EXEC divergence; WMMA requires EXEC all-ones).
  unsigned lane = threadIdx.x & 31u;
  bool lh = lane < 16u;
  int srcLo = (int)(lane & 15u);   // pull group-0 data (points = lanes 0..15)
  int srcHi = srcLo | 16;          // pull group-1 data (points = lanes 16..31)

  v2f a, b;
  b.x = 1.0f;
  b.y = 1.0f;
  v8f acc0 = {};
  v8f acc1 = {};
  float t;

  // group 0, K-chunk 0 (channels 0..3)
  t = __shfl(g2, srcLo, 32); a.x = lh ? g0 : t;
  t = __shfl(g3, srcLo, 32); a.y = lh ? g1 : t;
  acc0 = __builtin_amdgcn_wmma_f32_16x16x4_f32(false, a, false, b, (short)0,
                                               acc0, false, false);
  // group 0, K-chunk 1 (channels 4..7)
  t = __shfl(g6, srcLo, 32); a.x = lh ? g4 : t;
  t = __shfl(g7, srcLo, 32); a.y = lh ? g5 : t;
  acc0 = __builtin_amdgcn_wmma_f32_16x16x4_f32(false, a, false, b, (short)0,
                                               acc0, false, false);
  // group 1, K-chunk 0
  t = __shfl(g0, srcHi, 32); a.x = lh ? t : g2;
  t = __shfl(g1, srcHi, 32); a.y = lh ? t : g3;
  acc1 = __builtin_amdgcn_wmma_f32_16x16x4_f32(false, a, false, b, (short)0,
                                               acc1, false, false);
  // group 1, K-chunk 1
  t = __shfl(g4, srcHi, 32); a.x = lh ? t : g6;
  t = __shfl(g5, srcHi, 32); a.y = lh ? t : g7;
  acc1 = __builtin_amdgcn_wmma_f32_16x16x4_f32(false, a, false, b, (short)0,
                                               acc1, false, false);

  // D layout: acc[r], lanes 0-15 -> sigma_r ; lanes 16-31 -> sigma_{8+r}.
  // Map so all 32 lanes emit distinct points in one full-wave store.
  int s = (int)(lane & 7u);
  float sg0 = pick8(acc0, s);
  float sg1 = pick8(acc1, s);
  float sig = (lane & 8u) ? sg1 : sg0;
  int pidx = ((lane & 8u) ? 16 : 0) + (int)((lane >> 4) & 1u) * 8 + s;
  int waveBase = id & ~31;
  out[waveBase + pidx] = __expf(sig);
}

// ---------------- launcher ----------------
extern "C" void kernel_launch(void* const* d_in, const int* in_sizes, int n_in,
                              void* d_out, int out_size, void* d_ws,
                              size_t ws_size, hipStream_t stream) {
  const float* pts  = (const float*)d_in[0];
  const float* pxy  = (const float*)d_in[1];
  const float* pxz  = (const float*)d_in[2];
  const float* pyz  = (const float*)d_in[3];
  const float* w1   = (const float*)d_in[4];
  const float* w2   = (const float*)d_in[5];
  const float* aabb = (const float*)d_in[6];
  float* out = (float*)d_out;

  // workspace layout: [0,256): v (8 floats); [256, 256+6MB): repacked planes
  float* v = (float*)d_ws;
  float* planes = (float*)d_ws + 64;

  int npts = in_sizes[0] / 3;  // 16384 * 256

  kplane_compute_v<<<1, 32, 0, stream>>>(w1, w2, v);
  kplane_repack<<<(3 * RES * RES) / 256, 256, 0, stream>>>(pxy, pxz, pyz, v,
                                                           planes);
  kplane_main<<<npts / 256, 256, 0, stream>>>(pts, planes, aabb, out);
}